// Stage1Criterion_14800457302651
// MI455X (gfx1250) — compile-verified
//
#include <hip/hip_runtime.h>
#include <hip/hip_bf16.h>
#include <math.h>

typedef __attribute__((ext_vector_type(2))) float v2f;
typedef __attribute__((ext_vector_type(8))) float v8f;
typedef unsigned int u32;
typedef u32 u32x4 __attribute__((ext_vector_type(4)));
typedef u32 u32x8 __attribute__((ext_vector_type(8)));

// ---------------------------------------------------------------- reductions
__device__ __forceinline__ float wave_sum(float v) {
#pragma unroll
  for (int o = 16; o > 0; o >>= 1) v += __shfl_down(v, o, 32);
  return v;
}
__device__ __forceinline__ float wave_max(float v) {
#pragma unroll
  for (int o = 16; o > 0; o >>= 1) v = fmaxf(v, __shfl_down(v, o, 32));
  return v;
}

// ---------------------------------------------------------------- TDM 2-D tile load
// One Tensor-Data-Mover descriptor: copy a tile_h x tile_w (elements, f32) tile
// whose top-left is at `gaddr`, row stride `row_stride` elements, into LDS at
// byte offset `lds_off` (rows packed contiguously). Tracked by TENSORcnt.
__device__ __forceinline__ void tdm_load_2d(u32 lds_off, const void* gaddr,
                                            u32 tile_w, u32 tile_h,
                                            u32 row_stride, u32 tensor_w, u32 tensor_h) {
  unsigned long long ga = (unsigned long long)(size_t)gaddr;
  u32x4 g0;
  g0[0] = 1u;                                              // count=1 (valid user desc)
  g0[1] = lds_off;                                         // lds_addr (bytes)
  g0[2] = (u32)(ga & 0xFFFFFFFFu);                         // global_addr[31:0]
  g0[3] = (u32)((ga >> 32) & 0x01FFFFFFu) | (2u << 30);    // global_addr[56:32] | type=2
  u32x8 g1;
  g1[0] = (2u << 16);                                      // wg_mask=0, data_size=4B
  g1[1] = (tensor_w & 0xFFFFu) << 16;                      // tensor_dim0[15:0]
  g1[2] = ((tensor_w >> 16) & 0xFFFFu) | ((tensor_h & 0xFFFFu) << 16);
  g1[3] = ((tensor_h >> 16) & 0xFFFFu) | ((tile_w & 0xFFFFu) << 16);  // tile_dim0
  g1[4] = (tile_h & 0xFFFFu);                              // tile_dim1 (tile_dim2=0)
  g1[5] = row_stride;                                      // tensor_dim0_stride[31:0]
  g1[6] = 0u;                                              // stride hi / dim1_stride lo
  g1[7] = 0u;
  u32x4 gz = {0u, 0u, 0u, 0u};                             // groups 2/3: zero (2-D tile)
  asm volatile("tensor_load_to_lds %0, %1, %2, %3"
               :: "s"(g0), "s"(g1), "s"(gz), "s"(gz)
               : "memory");
}

__device__ __forceinline__ u32 lds_off_of(const void* p) {
  return (u32)(size_t)p;  // LDS aperture: addr[31:0] is the LDS byte offset
}

// ---------------------------------------------------------------- utilities
__global__ void zero_kernel(float* __restrict__ p, int n) {
  int i = blockIdx.x * blockDim.x + threadIdx.x;
  if (i < n) p[i] = 0.f;
}

__global__ void scale_kernel(float* __restrict__ p, int n, float f) {
  int i = blockIdx.x * blockDim.x + threadIdx.x;
  if (i < n) p[i] *= f;
}

// ---------------------------------------------------------------- cross entropy
__global__ void xe_kernel(const float* __restrict__ outputs,
                          const int* __restrict__ targets,
                          float* __restrict__ scalars, int C) {
  __shared__ float red[32];
  __shared__ float bc;
  const int row = blockIdx.x;
  const float* x = outputs + (size_t)row * C;
  const int tid = threadIdx.x, lane = tid & 31, wv = tid >> 5;
  const int nw = blockDim.x >> 5;

  float m = -3.4e38f;
  for (int j = tid; j < C; j += blockDim.x) m = fmaxf(m, x[j]);
  m = wave_max(m);
  if (lane == 0) red[wv] = m;
  __syncthreads();
  if (tid == 0) {
    float v = -3.4e38f;
    for (int w = 0; w < nw; ++w) v = fmaxf(v, red[w]);
    bc = v;
  }
  __syncthreads();
  m = bc;

  float s = 0.f;
  for (int j = tid; j < C; j += blockDim.x) s += __expf(x[j] - m);
  s = wave_sum(s);
  __syncthreads();
  if (lane == 0) red[wv] = s;
  __syncthreads();
  if (tid == 0) {
    float tot = 0.f;
    for (int w = 0; w < nw; ++w) tot += red[w];
    const int t = targets[row];
    atomicAdd(&scalars[0], -(x[t] - m - __logf(tot)));
  }
}

// ---------------------------------------------------------------- column mean
__global__ void mu_kernel(const float* __restrict__ emb, float* __restrict__ mu_sum,
                          int N, int D) {
  const int rows_per = (N + gridDim.x - 1) / gridDim.x;
  const int r0 = blockIdx.x * rows_per;
  const int r1 = min(N, r0 + rows_per);
  for (int c = threadIdx.x; c < D; c += blockDim.x) {
    float s = 0.f;
    for (int r = r0; r < r1; ++r) s += emb[(size_t)r * D + c];
    atomicAdd(&mu_sum[c], s);
  }
}

// ---------------------------------------------------------------- raw SYRK (WMMA + TDM)
// cov_sum += E^T E over a split-K chunk (rank-1 mu correction applied later).
// 8 waves; each owns two 16x16 WMMA subtiles of the 64x64 macro tile.
// K-slabs (16 x 64) are DMA'd into LDS by the Tensor Data Mover, double-buffered.
__global__ void cov_kernel(const float* __restrict__ emb,
                           float* __restrict__ cov_sum, int N, int D, int kchunk) {
  __shared__ float Ei[2][16][64];
  __shared__ float Ej[2][16][64];
  const int i0 = blockIdx.x * 64;
  const int j0 = blockIdx.y * 64;
  const int kstart = blockIdx.z * kchunk;
  const int kend = min(N, kstart + kchunk);
  const int nslab = (kend - kstart + 15) >> 4;
  const int tid = threadIdx.x;
  const int lane = tid & 31;
  const int wave = tid >> 5;

  const int s0 = wave * 2, s1 = wave * 2 + 1;
  const int mi0 = s0 >> 2, nj0 = s0 & 3;
  const int mi1 = s1 >> 2, nj1 = s1 & 3;
  const int m = lane & 15;
  const int khalf = (lane < 16) ? 0 : 2;

  v8f acc0 = {};
  v8f acc1 = {};

  if (wave == 0 && nslab > 0) {
    const int kb = kstart;
    const int krows = min(16, kend - kb);
    tdm_load_2d(lds_off_of(&Ei[0][0][0]), emb + (size_t)kb * D + i0, 64, krows,
                (u32)D, (u32)D, (u32)N);
    tdm_load_2d(lds_off_of(&Ej[0][0][0]), emb + (size_t)kb * D + j0, 64, krows,
                (u32)D, (u32)D, (u32)N);
  }

  for (int s = 0; s < nslab; ++s) {
    const int buf = s & 1;
    if (wave == 0) {
      if (s + 1 < nslab) {
        const int kb = kstart + (s + 1) * 16;
        const int krows = min(16, kend - kb);
        tdm_load_2d(lds_off_of(&Ei[buf ^ 1][0][0]), emb + (size_t)kb * D + i0, 64, krows,
                    (u32)D, (u32)D, (u32)N);
        tdm_load_2d(lds_off_of(&Ej[buf ^ 1][0][0]), emb + (size_t)kb * D + j0, 64, krows,
                    (u32)D, (u32)D, (u32)N);
        __builtin_amdgcn_s_wait_tensorcnt(2);  // slab s complete (in-order)
      } else {
        __builtin_amdgcn_s_wait_tensorcnt(0);
      }
    }
    __syncthreads();  // slab s visible to all waves
    // rare partial slab: zero stale tail rows, then re-sync
    const int krows = min(16, kend - (kstart + s * 16));
    if (krows < 16) {
      for (int idx = tid; idx < (16 - krows) * 64; idx += 256) {
        const int kk = krows + (idx >> 6), c = idx & 63;
        Ei[buf][kk][c] = 0.f;
        Ej[buf][kk][c] = 0.f;
      }
      __syncthreads();
    }
#pragma unroll
    for (int ks = 0; ks < 4; ++ks) {
      const int kk = ks * 4 + khalf;
      v2f a0, b0, a1, b1;
      a0.x = Ei[buf][kk][mi0 * 16 + m];  a0.y = Ei[buf][kk + 1][mi0 * 16 + m];
      b0.x = Ej[buf][kk][nj0 * 16 + m];  b0.y = Ej[buf][kk + 1][nj0 * 16 + m];
      acc0 = __builtin_amdgcn_wmma_f32_16x16x4_f32(false, a0, false, b0,
                                                   (short)0, acc0, false, false);
      a1.x = Ei[buf][kk][mi1 * 16 + m];  a1.y = Ei[buf][kk + 1][mi1 * 16 + m];
      b1.x = Ej[buf][kk][nj1 * 16 + m];  b1.y = Ej[buf][kk + 1][nj1 * 16 + m];
      acc1 = __builtin_amdgcn_wmma_f32_16x16x4_f32(false, a1, false, b1,
                                                   (short)0, acc1, false, false);
    }
    __syncthreads();  // all waves done with buf before it is re-DMA'd
  }

  const int mb = (lane < 16) ? 0 : 8;
#pragma unroll
  for (int r = 0; r < 8; ++r) {
    atomicAdd(&cov_sum[(size_t)(i0 + mi0 * 16 + mb + r) * D + (j0 + nj0 * 16 + m)], acc0[r]);
    atomicAdd(&cov_sum[(size_t)(i0 + mi1 * 16 + mb + r) * D + (j0 + nj1 * 16 + m)], acc1[r]);
  }
}

// ---------------------------------------------------------------- Gauss-Jordan inverse
// cov = (raw - N*mu*mu^T)/(N-1); SPD -> diagonal pivoting. One thread per row.
__global__ void gj_kernel(const float* __restrict__ covsum, const float* __restrict__ mu,
                          float* __restrict__ aug, int D, float fN, float inv_nm1) {
  __shared__ float piv[1024];  // 2*D, D<=512
  const int tid = threadIdx.x;
  const int W = 2 * D;
  float* myrow = aug + (size_t)tid * W;
  const float* crow = covsum + (size_t)tid * D;
  const float mui = mu[tid] * fN;
  for (int j = 0; j < D; ++j) myrow[j] = (crow[j] - mui * mu[j]) * inv_nm1;
  for (int j = 0; j < D; ++j) myrow[D + j] = (j == tid) ? 1.f : 0.f;
  __syncthreads();
  for (int k = 0; k < D; ++k) {
    const float p = aug[(size_t)k * W + k];
    const float invp = 1.f / p;
    __syncthreads();
    for (int j = tid; j < W; j += D) {
      const float v = aug[(size_t)k * W + j] * invp;
      aug[(size_t)k * W + j] = v;
      piv[j] = v;
    }
    __syncthreads();
    if (tid != k) {
      const float f = myrow[k];
      for (int j = 0; j < W; ++j) myrow[j] -= f * piv[j];
    }
    __syncthreads();
  }
}

// ---------------------------------------------------------------- weight row-normalize
__global__ void wnorm_kernel(const float* __restrict__ w, float* __restrict__ what, int D) {
  __shared__ float red[32];
  __shared__ float bc;
  const int row = blockIdx.x;
  const float* x = w + (size_t)row * D;
  const int tid = threadIdx.x, lane = tid & 31, wv = tid >> 5;
  const int nw = blockDim.x >> 5;
  float s = 0.f;
  for (int j = tid; j < D; j += blockDim.x) { float v = x[j]; s += v * v; }
  s = wave_sum(s);
  if (lane == 0) red[wv] = s;
  __syncthreads();
  if (tid == 0) {
    float tot = 0.f;
    for (int w2 = 0; w2 < nw; ++w2) tot += red[w2];
    bc = rsqrtf(tot);
  }
  __syncthreads();
  const float inv = bc;
  for (int j = tid; j < D; j += blockDim.x) what[(size_t)row * D + j] = x[j] * inv;
}

// ---------------------------------------------------------------- Mahalanobis (WMMA + TDM)
// 128 threads = 4 waves, 64 rows per block. Centered rows staged transposed in
// LDS (pad 65 -> conflict-free); Sinv column panels (512x16) DMA'd by the TDM.
// Y = T @ Sinv tile-by-tile via WMMA, s += sum T.*Y, wave-reduce, atomicAdd.
__global__ void maha_kernel(const float* __restrict__ X, int R,
                            const float* __restrict__ mu,
                            const float* __restrict__ sinv, int sld,
                            float* __restrict__ out_scalar, int D) {
  __shared__ float Tt[512][65];   // [k][row], centered (D<=512, 64 rows used)
  __shared__ float Sp[512][16];   // Sinv panel [k][n]
  const int tid = threadIdx.x;
  const int lane = tid & 31, wave = tid >> 5;
  const int rbase = blockIdx.x * 64;

  for (int idx = tid; idx < 64 * D; idx += blockDim.x) {
    const int col = idx % D;
    const int row = idx / D;
    const int grow = rbase + row;
    float v = 0.f;
    if (grow < R) v = X[(size_t)grow * D + col] - mu[col];
    Tt[col][row] = v;
  }
  __syncthreads();

  const int m = lane & 15;
  const int khalf = (lane < 16) ? 0 : 2;
  const int mb = (lane < 16) ? 0 : 8;
  const int rloc = wave * 16;
  float s = 0.f;

  for (int nj = 0; nj < D; nj += 16) {
    if (wave == 0) {
      tdm_load_2d(lds_off_of(&Sp[0][0]), sinv + nj, 16, (u32)D,
                  (u32)sld, (u32)sld, (u32)D);
      __builtin_amdgcn_s_wait_tensorcnt(0);
    }
    __syncthreads();  // panel ready
    v8f acc = {};
    for (int k = 0; k < D; k += 4) {
      const int kk = k + khalf;
      v2f a, b;
      a.x = Tt[kk][rloc + m];      a.y = Tt[kk + 1][rloc + m];
      b.x = Sp[kk][m];             b.y = Sp[kk + 1][m];
      acc = __builtin_amdgcn_wmma_f32_16x16x4_f32(false, a, false, b,
                                                  (short)0, acc, false, false);
    }
#pragma unroll
    for (int r = 0; r < 8; ++r) {
      s += acc[r] * Tt[nj + m][rloc + mb + r];
    }
    __syncthreads();  // all waves done with Sp before next DMA
  }
  s = wave_sum(s);
  if (lane == 0) atomicAdd(out_scalar, s);
}

// ---------------------------------------------------------------- finalize
__global__ void finalize_kernel(const float* __restrict__ scalars, float* __restrict__ out,
                                int B, int N, int C) {
  const float xe = scalars[0] / (float)B;
  const float md = scalars[1] / (float)N;
  const float mw = scalars[2] / (float)C;
  out[0] = xe + (mw - md) / (md * md * md);
}

// ---------------------------------------------------------------- launch
extern "C" void kernel_launch(void* const* d_in, const int* in_sizes, int n_in,
                              void* d_out, int out_size, void* d_ws, size_t ws_size,
                              hipStream_t stream) {
  const float* outputs = (const float*)d_in[0];
  const int*   targets = (const int*)d_in[1];
  const float* weights = (const float*)d_in[2];
  const float* emb     = (const float*)d_in[3];

  const int B = in_sizes[1];
  const int C = in_sizes[0] / B;
  const int D = in_sizes[2] / C;
  const int N = in_sizes[3] / D;

  float* ws      = (float*)d_ws;
  float* mu      = ws;                         // D floats
  float* scalars = ws + D;                     // [0]=xe_sum [1]=emb_dist [2]=w_dist
  float* cov     = ws + 1024;                  // D*D raw E^T E sums
  float* aug     = cov + (size_t)D * D;        // D * 2D augmented GJ
  float* what    = aug + (size_t)D * 2 * D;    // C*D normalized weights

  const int zn = 1024 + D * D;
  zero_kernel<<<(zn + 255) / 256, 256, 0, stream>>>(ws, zn);

  xe_kernel<<<B, 256, 0, stream>>>(outputs, targets, scalars, C);

  mu_kernel<<<512, 256, 0, stream>>>(emb, mu, N, D);
  scale_kernel<<<(D + 255) / 256, 256, 0, stream>>>(mu, D, 1.0f / (float)N);

  const int Z = 16;
  const int kchunk = ((N + Z * 16 - 1) / (Z * 16)) * 16;  // multiple of 16
  dim3 cgrid(D / 64, D / 64, Z);
  cov_kernel<<<cgrid, 256, 0, stream>>>(emb, cov, N, D, kchunk);

  gj_kernel<<<1, D, 0, stream>>>(cov, mu, aug, D, (float)N, 1.0f / (float)(N - 1));

  wnorm_kernel<<<C, 256, 0, stream>>>(weights, what, D);

  const float* sinv = aug + D;   // right half of augmented matrix
  const int sld = 2 * D;

  maha_kernel<<<(N + 63) / 64, 128, 0, stream>>>(emb, N, mu, sinv, sld, scalars + 1, D);
  maha_kernel<<<(C + 63) / 64, 128, 0, stream>>>(what, C, mu, sinv, sld, scalars + 2, D);

  finalize_kernel<<<1, 1, 0, stream>>>(scalars, (float*)d_out, B, N, C);
}